// MyModule_24026047054338
// MI455X (gfx1250) — compile-verified
//
#include <hip/hip_runtime.h>

typedef __attribute__((ext_vector_type(2))) float v2f;
typedef __attribute__((ext_vector_type(4))) float v4f;
typedef __attribute__((ext_vector_type(8))) float v8f;

#define M_DIM 4096
#define N_DIM 1024
#define K_DIM 1024
#define MASK_N (512 * 512 * 512)

// ---------------------------------------------------------------------------
// GEMM: out[b,o] = sum_k input[b,k] * weight[o,k] + bias[o]
// One wave computes a 16x64 strip: 1 A-fragment feeds 4 v_wmma_f32_16x16x4_f32
// per K-step. Accumulators live in 4x8 VGPRs (16x16 f32 C/D layout).
// ---------------------------------------------------------------------------
__global__ __launch_bounds__(256) void gemm_wmma_f32(
    const float* __restrict__ A,     // [4096,1024] row-major
    const float* __restrict__ W,     // [1024(out),1024(in)] row-major
    const float* __restrict__ bias,  // [1024]
    float* __restrict__ out)         // [4096,1024]
{
    const int lane   = threadIdx.x & 31;
    const int waveId = blockIdx.x * 8 + (threadIdx.x >> 5); // 4096 waves total
    const int mtile  = waveId >> 4;   // 0..255  (M / 16)
    const int nstrip = waveId & 15;   // 0..15   (N / 64)
    const int m0 = mtile * 16;
    const int n0 = nstrip * 64;

    const int lr = lane & 15;   // row (A) / col (B) index within 16
    const int hi = lane >> 4;   // selects K-pair {0,1} vs {2,3}

    v8f acc0 = {}, acc1 = {}, acc2 = {}, acc3 = {};

    // A fragment: lane holds A[m0+lr][kk + 2*hi + {0,1}]  (8B-aligned pair)
    const float* aP  = A + (size_t)(m0 + lr) * K_DIM + hi * 2;
    // B fragment: lane holds W[n0+t*16+lr][kk + 2*hi + {0,1}]
    const float* bP0 = W + (size_t)(n0 +  0 + lr) * K_DIM + hi * 2;
    const float* bP1 = W + (size_t)(n0 + 16 + lr) * K_DIM + hi * 2;
    const float* bP2 = W + (size_t)(n0 + 32 + lr) * K_DIM + hi * 2;
    const float* bP3 = W + (size_t)(n0 + 48 + lr) * K_DIM + hi * 2;

    #pragma unroll 4
    for (int kk = 0; kk < K_DIM; kk += 4) {
        v2f a  = *(const v2f*)(aP  + kk);
        v2f b0 = *(const v2f*)(bP0 + kk);
        v2f b1 = *(const v2f*)(bP1 + kk);
        v2f b2 = *(const v2f*)(bP2 + kk);
        v2f b3 = *(const v2f*)(bP3 + kk);
        // (neg_a, A, neg_b, B, c_mod, C, reuse_a, reuse_b)
        acc0 = __builtin_amdgcn_wmma_f32_16x16x4_f32(false, a, false, b0, (short)0, acc0, false, false);
        acc1 = __builtin_amdgcn_wmma_f32_16x16x4_f32(false, a, false, b1, (short)0, acc1, false, false);
        acc2 = __builtin_amdgcn_wmma_f32_16x16x4_f32(false, a, false, b2, (short)0, acc2, false, false);
        acc3 = __builtin_amdgcn_wmma_f32_16x16x4_f32(false, a, false, b3, (short)0, acc3, false, false);
    }

    // C/D layout: VGPR r, lanes 0-15 -> (M = r,   N = lane)
    //                     lanes 16-31 -> (M = r+8, N = lane-16)
    const int rowOff = hi * 8;
    {
        const int n = n0 + 0 * 16 + lr;
        const float bv = bias[n];
        #pragma unroll
        for (int r = 0; r < 8; ++r)
            out[(size_t)(m0 + r + rowOff) * N_DIM + n] = acc0[r] + bv;
    }
    {
        const int n = n0 + 1 * 16 + lr;
        const float bv = bias[n];
        #pragma unroll
        for (int r = 0; r < 8; ++r)
            out[(size_t)(m0 + r + rowOff) * N_DIM + n] = acc1[r] + bv;
    }
    {
        const int n = n0 + 2 * 16 + lr;
        const float bv = bias[n];
        #pragma unroll
        for (int r = 0; r < 8; ++r)
            out[(size_t)(m0 + r + rowOff) * N_DIM + n] = acc2[r] + bv;
    }
    {
        const int n = n0 + 3 * 16 + lr;
        const float bv = bias[n];
        #pragma unroll
        for (int r = 0; r < 8; ++r)
            out[(size_t)(m0 + r + rowOff) * N_DIM + n] = acc3[r] + bv;
    }
}

// ---------------------------------------------------------------------------
// Deterministic 2-stage reduction: threshold = sum(out) / 4096
// ---------------------------------------------------------------------------
__global__ __launch_bounds__(256) void reduce_stage1(
    const float* __restrict__ x, float* __restrict__ partial)
{
    __shared__ float sm[256];
    const int base = blockIdx.x * 1024;
    float s = x[base + threadIdx.x]       + x[base + threadIdx.x + 256]
            + x[base + threadIdx.x + 512] + x[base + threadIdx.x + 768];
    sm[threadIdx.x] = s;
    __syncthreads();
    for (int off = 128; off > 0; off >>= 1) {
        if (threadIdx.x < (unsigned)off) sm[threadIdx.x] += sm[threadIdx.x + off];
        __syncthreads();
    }
    if (threadIdx.x == 0) partial[blockIdx.x] = sm[0];
}

__global__ __launch_bounds__(256) void reduce_stage2(float* __restrict__ ws)
{
    __shared__ float sm[256];
    float s = 0.0f;
    #pragma unroll
    for (int i = 0; i < 16; ++i) s += ws[threadIdx.x + 256 * i];
    sm[threadIdx.x] = s;
    __syncthreads();
    for (int off = 128; off > 0; off >>= 1) {
        if (threadIdx.x < (unsigned)off) sm[threadIdx.x] += sm[threadIdx.x + off];
        __syncthreads();
    }
    if (threadIdx.x == 0) ws[4096] = sm[0] * (1.0f / 4096.0f);
}

// ---------------------------------------------------------------------------
// Streaming mask compare: hi[i] = (mask[i] > threshold) ? 1.0f : 0.0f
// b128 loads/stores; ~1.07 GB of traffic -> HBM-bound (~46us at 23.3 TB/s)
// ---------------------------------------------------------------------------
__global__ __launch_bounds__(256) void mask_cmp(
    const float* __restrict__ mask, const float* __restrict__ thr,
    float* __restrict__ hiMask, int n4)
{
    const float t = thr[0];
    const int stride = gridDim.x * blockDim.x;
    for (int i = blockIdx.x * blockDim.x + threadIdx.x; i < n4; i += stride) {
        __builtin_prefetch(mask + (size_t)(i + stride) * 4, 0, 0);
        v4f m = *(const v4f*)(mask + (size_t)i * 4);
        v4f r;
        r.x = (m.x > t) ? 1.0f : 0.0f;
        r.y = (m.y > t) ? 1.0f : 0.0f;
        r.z = (m.z > t) ? 1.0f : 0.0f;
        r.w = (m.w > t) ? 1.0f : 0.0f;
        *(v4f*)(hiMask + (size_t)i * 4) = r;
    }
}

// ---------------------------------------------------------------------------
extern "C" void kernel_launch(void* const* d_in, const int* in_sizes, int n_in,
                              void* d_out, int out_size, void* d_ws, size_t ws_size,
                              hipStream_t stream)
{
    const float* input  = (const float*)d_in[0];  // [4096,1024]
    const float* mask   = (const float*)d_in[1];  // [512,512,512]
    const float* weight = (const float*)d_in[2];  // [1024,1024]
    const float* bias   = (const float*)d_in[3];  // [1024]

    float* out    = (float*)d_out;                    // first output, 4096*1024
    float* hiMask = out + (size_t)M_DIM * N_DIM;      // second output, 512^3
    float* ws     = (float*)d_ws;                     // [0..4095] partials, [4096] threshold

    // GEMM: 4096 waves (256 M-tiles x 16 N-strips), 8 waves per 256-thread block
    gemm_wmma_f32<<<512, 256, 0, stream>>>(input, weight, bias, out);

    // threshold = mean over batch of per-row sums = sum(out)/4096, deterministic order
    reduce_stage1<<<4096, 256, 0, stream>>>(out, ws);
    reduce_stage2<<<1, 256, 0, stream>>>(ws);

    // mask > threshold
    const int n4 = MASK_N / 4;
    mask_cmp<<<16384, 256, 0, stream>>>(mask, ws + 4096, hiMask, n4);
}